// IntermediateLossLayer_10471130268492
// MI455X (gfx1250) — compile-verified
//
#include <hip/hip_runtime.h>
#include <hip/hip_bf16.h>
#include <math.h>

// ---------------------------------------------------------------------------
// ARAP edge-rigidity loss, MI455X (gfx1250, wave32).
//
// loss = 0.5 * rigidity^2 * sum_e ( ||V[e0]-V[e1]|| - rest[e] )^2
//
// Roofline: ~240 MB single-use stream (E1 int64 pairs + rest_len) + 24 MB of
// V1 random gathers.  V1 fits in the 192 MB L2, so gathers are L2-resident
// *iff* the 240 MB stream doesn't evict it -> stream loads are issued with
// the CDNA5 non-temporal hint (TH=NT) via __builtin_nontemporal_load.
// HBM floor at 23.3 TB/s: ~10-12 us.  Compute (~0.12 GFLOP) is negligible.
//
// Reductions: one V_WMMA_F32_16X16X4_F32 against an all-ones B matrix does a
// full wave32 sum (lane-local 8-VGPR add + one xor-16 bpermute), then a
// fixed-order LDS block reduce and a fixed-order final pass -> bitwise
// deterministic (no float atomics).
// ---------------------------------------------------------------------------

typedef float     v2f   __attribute__((ext_vector_type(2)));
typedef float     v8f   __attribute__((ext_vector_type(8)));
typedef long long v2i64 __attribute__((ext_vector_type(2)));   // one b128 load

#define TPB  256       // 8 wave32 waves per workgroup
#define NBLK 4096      // 1,048,576 lanes -> ~12 edges per lane

// Full wave32 reduction via one f32 WMMA.
// A (16x4): VGPR0 = p, VGPR1 = 0  =>  A[m][0] = p_m (lanes 0-15),
//                                     A[m][2] = p_{m+16} (lanes 16-31).
// B (4x16) = all ones  =>  D[m][n] = p_m + p_{m+16} for every n.
// Lane L<16 holds D-rows 0..7 in its 8 dest VGPRs, lane L>=16 rows 8..15;
// summing them gives two half-sums, combined with one xor-16 shuffle.
// Requires EXEC == all ones (uniform control flow at call site).
__device__ __forceinline__ float wave_reduce_wmma(float p) {
    v2f a;  a[0] = p;    a[1] = 0.0f;
    v2f b;  b[0] = 1.0f; b[1] = 1.0f;
    v8f c = {};
    c = __builtin_amdgcn_wmma_f32_16x16x4_f32(
            /*neg_a=*/false, a, /*neg_b=*/false, b,
            /*c_mod=*/(short)0, c, /*reuse_a=*/false, /*reuse_b=*/false);
    float s = ((c[0] + c[1]) + (c[2] + c[3])) + ((c[4] + c[5]) + (c[6] + c[7]));
    s += __shfl_xor(s, 16, 32);
    return s;   // every lane holds the 32-lane total
}

// Deterministic block reduction: WMMA wave reduce -> LDS -> fixed-order sum.
__device__ __forceinline__ float block_reduce(float p, float* lds) {
    float w = wave_reduce_wmma(p);
    const int lane = threadIdx.x & 31;
    const int wid  = threadIdx.x >> 5;
    if (lane == 0) lds[wid] = w;
    __syncthreads();
    float s = 0.0f;
    if (threadIdx.x == 0) {
        #pragma unroll
        for (int i = 0; i < TPB / 32; ++i) s += lds[i];
    }
    return s;   // valid in thread 0 only
}

__global__ __launch_bounds__(TPB)
void arap_edge_partial(const float* __restrict__ V,     // [N,3] f32
                       const v2i64* __restrict__ E,     // [M,2] i64 pairs
                       const float* __restrict__ rest,  // [M]   f32
                       float*       __restrict__ partial,
                       int M)
{
    __shared__ float lds[TPB / 32];
    float acc = 0.0f;

    const int stride = (int)(gridDim.x * blockDim.x);
    for (int i = (int)(blockIdx.x * blockDim.x + threadIdx.x); i < M; i += stride) {
        // Single-use streams: non-temporal so they don't evict V1 from L2.
        const v2i64 e = __builtin_nontemporal_load(&E[i]);   // b128, TH=NT
        const float r = __builtin_nontemporal_load(&rest[i]);// b32,  TH=NT
        const float* pa = V + 3ll * e.x;                     // gathers: L2-resident
        const float* pb = V + 3ll * e.y;
        const float dx = pa[0] - pb[0];
        const float dy = pa[1] - pb[1];
        const float dz = pa[2] - pb[2];
        const float len  = sqrtf(fmaf(dx, dx, fmaf(dy, dy, fmaf(dz, dz, 1e-12f))));
        const float diff = len - r;
        acc = fmaf(diff, diff, acc);
    }
    // Control flow reconverges here: EXEC all ones before the WMMA reduce.
    const float s = block_reduce(acc, lds);
    if (threadIdx.x == 0) partial[blockIdx.x] = s;
}

__global__ __launch_bounds__(TPB)
void arap_final_reduce(const float* __restrict__ partial, int n,
                       const float* __restrict__ rig2,
                       float*       __restrict__ out)
{
    __shared__ float lds[TPB / 32];
    float acc = 0.0f;
    for (int i = (int)threadIdx.x; i < n; i += TPB) acc += partial[i];
    const float s = block_reduce(acc, lds);
    if (threadIdx.x == 0) out[0] = 0.5f * rig2[0] * s;
}

extern "C" void kernel_launch(void* const* d_in, const int* in_sizes, int n_in,
                              void* d_out, int out_size, void* d_ws, size_t ws_size,
                              hipStream_t stream) {
    const float* V    = (const float*)d_in[0];       // V1: [N,3] f32
    const v2i64* E    = (const v2i64*)d_in[1];       // E1: [M,2] i64
    const float* rest = (const float*)d_in[2];       // rest_len: [M] f32
    const float* rig2 = (const float*)d_in[3];       // rigidity2 scalar (f32)
    // d_in[4] = src_param_id: unused (rest lengths passed explicitly).

    const int M = in_sizes[2];                       // N_EDGES from rest_len
    float* partial = (float*)d_ws;                   // NBLK floats scratch

    arap_edge_partial<<<NBLK, TPB, 0, stream>>>(V, E, rest, partial, M);
    arap_final_reduce<<<1, TPB, 0, stream>>>(partial, NBLK, rig2, (float*)d_out);
}